// SemanticBatchHardXBM_33268816675423
// MI455X (gfx1250) — compile-verified
//
#include <hip/hip_runtime.h>
#include <math.h>

// Problem constants (from reference)
#define N_PTS   8192
#define WV_DIM  300
#define D_DIM   128
#define BIG_F   1e8f
#define EPS_TRM (128.0f * 1e-16f)   // D * EPS added under the sqrt

typedef float v2f __attribute__((ext_vector_type(2)));
typedef float v8f __attribute__((ext_vector_type(8)));

__device__ __forceinline__ v2f ld2(const float* p) {
    return *(const v2f*)p;   // 8-byte aligned global load (global_load_b64)
}

// ---------------------------------------------------------------------------
// Kernel 1: x2[j] = ||x_j||^2   (trivial, bandwidth ~4MB, L2-resident after)
// ---------------------------------------------------------------------------
__global__ void x2_kernel(const float* __restrict__ x, float* __restrict__ x2) {
    int i = blockIdx.x * blockDim.x + threadIdx.x;
    if (i < N_PTS) {
        const float* row = x + (size_t)i * D_DIM;
        float s = 0.f;
        #pragma unroll 8
        for (int d = 0; d < D_DIM; ++d) s = fmaf(row[d], row[d], s);
        x2[i] = s;
    }
}

// ---------------------------------------------------------------------------
// Kernel 2: one wave per 16-row stripe.
//   sem_org = wv @ W^T + b  via V_WMMA_F32_16X16X4_F32 (K=300 = 75 steps)
//   sem     = alpha*sem_org + (1-alpha)*x   (stored to d_out sem region)
//   sem2[i] = ||sem_i||^2                    (width-16 shfl reduction)
// C-tile layout: lane = d-col (lane&15), VGPR r = row r + 8*(lane>=16).
// ---------------------------------------------------------------------------
__global__ __launch_bounds__(32) void sem_fc_kernel(
    const float* __restrict__ x,  const float* __restrict__ wv,
    const float* __restrict__ alpha, const float* __restrict__ W,
    const float* __restrict__ b,  float* __restrict__ sem_out,
    float* __restrict__ sem2) {
    const int lane = threadIdx.x;
    const int col  = lane & 15;    // column within 16-wide tile
    const int hi   = lane >> 4;    // half-wave select (K pair / row+8)
    const int i0   = blockIdx.x * 16;

    float alph[8];
    #pragma unroll
    for (int r = 0; r < 8; ++r) alph[r] = alpha[i0 + r + 8 * hi];

    float nrm[8];
    #pragma unroll
    for (int r = 0; r < 8; ++r) nrm[r] = 0.f;

    const float* arow = wv + (size_t)(i0 + col) * WV_DIM + 2 * hi;

    for (int dt = 0; dt < D_DIM / 16; ++dt) {
        const int d = dt * 16 + col;
        const float* brow = W + (size_t)d * WV_DIM + 2 * hi;

        v8f c = {};
        for (int k = 0; k < WV_DIM; k += 4) {
            v2f a  = ld2(arow + k);
            v2f bb = ld2(brow + k);
            c = __builtin_amdgcn_wmma_f32_16x16x4_f32(
                    false, a, false, bb, (short)0, c, false, false);
        }

        const float bd = b[d];
        #pragma unroll
        for (int r = 0; r < 8; ++r) {
            const int i = i0 + r + 8 * hi;
            const float so = c[r] + bd;
            const float xv = x[(size_t)i * D_DIM + d];
            const float s  = alph[r] * so + (1.f - alph[r]) * xv;
            sem_out[(size_t)i * D_DIM + d] = s;
            nrm[r] = fmaf(s, s, nrm[r]);
        }
    }

    // reduce ||sem_i||^2 across the 16 lanes of each half-wave
    #pragma unroll
    for (int r = 0; r < 8; ++r) {
        float v = nrm[r];
        for (int off = 8; off >= 1; off >>= 1)
            v += __shfl_xor(v, off, 16);
        if (col == 0) sem2[i0 + r + 8 * hi] = v;
    }
}

// ---------------------------------------------------------------------------
// Kernel 3: fused pairwise-distance + batch-hard reduction.
// One wave per 16-row i-stripe; A fragments (sem rows, 32 k-steps) are
// preloaded into registers and reused across all 512 j-tiles. Inner loop:
// one b64 load + one f32 WMMA per k-step; epilogue fuses sqrt, label masks,
// running max/min. Final width-16 shfl tree + stable softplus.
// ---------------------------------------------------------------------------
__global__ __launch_bounds__(32) void pair_kernel(
    const float* __restrict__ x,    const int* __restrict__ label,
    const float* __restrict__ sem,  const float* __restrict__ sem2,
    const float* __restrict__ x2,   float* __restrict__ diff) {
    const int lane = threadIdx.x;
    const int col  = lane & 15;
    const int hi   = lane >> 4;
    const int i0   = blockIdx.x * 16;

    // Preload all 32 A fragments for this stripe (invariant over j)
    v2f afrag[32];
    const float* arow = sem + (size_t)(i0 + col) * D_DIM + 2 * hi;
    #pragma unroll
    for (int ks = 0; ks < 32; ++ks) afrag[ks] = ld2(arow + 4 * ks);

    int   labi[8];
    float s2i[8];
    #pragma unroll
    for (int r = 0; r < 8; ++r) {
        labi[r] = label[i0 + r + 8 * hi];
        s2i[r]  = sem2[i0 + r + 8 * hi];
    }

    float fp[8], cn[8];
    #pragma unroll
    for (int r = 0; r < 8; ++r) { fp[r] = 0.f; cn[r] = 3.4e38f; }

    for (int jt = 0; jt < N_PTS / 16; ++jt) {
        const int   j    = jt * 16 + col;
        const int   labj = label[j];
        const float x2j  = x2[j];
        const float* brow = x + (size_t)j * D_DIM + 2 * hi;

        v8f c = {};
        #pragma unroll
        for (int ks = 0; ks < 32; ++ks) {
            v2f bb = ld2(brow + 4 * ks);
            c = __builtin_amdgcn_wmma_f32_16x16x4_f32(
                    false, afrag[ks], false, bb, (short)0, c, false, false);
        }

        #pragma unroll
        for (int r = 0; r < 8; ++r) {
            const int i  = i0 + r + 8 * hi;
            float sq = s2i[r] + x2j - 2.f * c[r];
            sq = sq > 0.f ? sq : 0.f;
            const float dist = sqrtf(sq + EPS_TRM);
            const bool same = (labi[r] == labj);
            const bool pos  = same && (i != j);
            fp[r] = fmaxf(fp[r], pos ? dist : 0.f);
            cn[r] = fminf(cn[r], same ? dist + BIG_F : dist);
        }
    }

    // width-16 max/min trees, then softplus(fp - cn)
    #pragma unroll
    for (int r = 0; r < 8; ++r) {
        float a = fp[r], m = cn[r];
        for (int off = 8; off >= 1; off >>= 1) {
            a = fmaxf(a, __shfl_xor(a, off, 16));
            m = fminf(m, __shfl_xor(m, off, 16));
        }
        if (col == 0) {
            const float z  = a - m;
            const float sp = (z > 20.f) ? z : log1pf(expf(z));  // stable softplus
            diff[i0 + r + 8 * hi] = sp;
        }
    }
}

// ---------------------------------------------------------------------------
extern "C" void kernel_launch(void* const* d_in, const int* in_sizes, int n_in,
                              void* d_out, int out_size, void* d_ws, size_t ws_size,
                              hipStream_t stream) {
    const float* x     = (const float*)d_in[0];
    const float* wv    = (const float*)d_in[1];
    const int*   label = (const int*)  d_in[2];
    const float* alpha = (const float*)d_in[3];
    const float* W     = (const float*)d_in[4];
    const float* b     = (const float*)d_in[5];

    float* diff = (float*)d_out;          // [N]
    float* sem  = diff + N_PTS;           // [N, D] — second output, also reused as GEMM input

    float* x2   = (float*)d_ws;           // [N] scratch
    float* sem2 = x2 + N_PTS;             // [N] scratch

    hipLaunchKernelGGL(x2_kernel, dim3(N_PTS / 256), dim3(256), 0, stream, x, x2);
    hipLaunchKernelGGL(sem_fc_kernel, dim3(N_PTS / 16), dim3(32), 0, stream,
                       x, wv, alpha, W, b, sem, sem2);
    hipLaunchKernelGGL(pair_kernel, dim3(N_PTS / 16), dim3(32), 0, stream,
                       x, label, sem, sem2, x2, diff);
}